// Conv_16930761081032
// MI455X (gfx1250) — compile-verified
//
#include <hip/hip_runtime.h>
#include <hip/hip_bf16.h>

typedef __attribute__((ext_vector_type(2))) float v2f;
typedef __attribute__((ext_vector_type(8))) float v8f;

#define CH 64
#define BN_EPS 1e-5f

// ---------------- 1) zero the workspace regions ----------------
__global__ void zero_kernel(float* p, long long n) {
    long long i = (long long)blockIdx.x * blockDim.x + threadIdx.x;
    long long stride = (long long)gridDim.x * blockDim.x;
    for (; i < n; i += stride) p[i] = 0.0f;
}

// ---------------- 2) scatter-add: sum[tgt][c] += x[src][c] ----------------
__global__ void scatter_kernel(const float* __restrict__ x,
                               const int* __restrict__ src,
                               const int* __restrict__ tgt,
                               float* __restrict__ sum,
                               float* __restrict__ cnt,
                               int nedges) {
    long long g = (long long)blockIdx.x * blockDim.x + threadIdx.x;
    int e = (int)(g >> 6);
    int c = (int)(g & 63);
    if (e >= nedges) return;
    int s = src[e];
    int t = tgt[e];
    unsafeAtomicAdd(&sum[(size_t)t * CH + c], x[(size_t)s * CH + c]);
    if (c == 0) unsafeAtomicAdd(&cnt[t], 1.0f);
}

// ---------------- 3) agg = sum / max(cnt,1) in place; channel sums ----------------
__global__ void stats_kernel(float* __restrict__ sum,
                             const float* __restrict__ cnt,
                             float* __restrict__ chanSum,
                             float* __restrict__ chanSumSq,
                             int nnodes) {
    int stride = gridDim.x * blockDim.x;      // multiple of 64 by construction
    int idx0 = blockIdx.x * blockDim.x + threadIdx.x;
    int c = idx0 & 63;                        // constant per thread across strides
    float s = 0.0f, s2 = 0.0f;
    int total = nnodes * CH;
    for (int idx = idx0; idx < total; idx += stride) {
        int n = idx >> 6;
        float inv = 1.0f / fmaxf(cnt[n], 1.0f);
        float a = sum[idx] * inv;
        sum[idx] = a;                          // sum buffer becomes agg in place
        s += a;
        s2 += a * a;
    }
    unsafeAtomicAdd(&chanSum[c], s);
    unsafeAtomicAdd(&chanSumSq[c], s2);
}

// ---------------- 4) fold BN into the Linear: W' (packed pairs) and b' ----------------
// WpPK layout: WpPK[k2*64 + o] = { W[o][2k2]*scale[2k2], W[o][2k2+1]*scale[2k2+1] }
__global__ void prep_kernel(const float* __restrict__ gamma,
                            const float* __restrict__ beta,
                            const float* __restrict__ W,
                            const float* __restrict__ b,
                            const float* __restrict__ chanSum,
                            const float* __restrict__ chanSumSq,
                            float2* __restrict__ WpPK,
                            float* __restrict__ bp,
                            float invN) {
    __shared__ float scale[CH];
    __shared__ float shift[CH];
    int t = threadIdx.x;                      // 64 threads
    float mean = chanSum[t] * invN;
    float var  = chanSumSq[t] * invN - mean * mean;
    float sc   = gamma[t] * rsqrtf(var + BN_EPS);
    scale[t] = sc;
    shift[t] = beta[t] - mean * sc;
    __syncthreads();
    // thread t == output channel o
    float acc = b[t];
    #pragma unroll 4
    for (int cc = 0; cc < CH; ++cc) acc += shift[cc] * W[t * CH + cc];
    bp[t] = acc;
    #pragma unroll 4
    for (int k2 = 0; k2 < CH / 2; ++k2) {
        float2 w;
        w.x = W[t * CH + 2 * k2]     * scale[2 * k2];
        w.y = W[t * CH + 2 * k2 + 1] * scale[2 * k2 + 1];
        WpPK[k2 * CH + t] = w;
    }
}

// ---------------- 5) out = relu(agg @ W'^T + b') via V_WMMA_F32_16X16X4_F32 ----------------
// Block: 256 threads = 8 waves; wave w -> tile (rowTile = w>>2, colTile = w&3)
// Block covers 32 rows x 64 output channels. K=64 via 16 chained K=4 WMMAs.
__global__ void gemm_kernel(const float* __restrict__ agg,
                            const float2* __restrict__ WpPK,
                            const float* __restrict__ bp,
                            float* __restrict__ out,
                            int nrows) {
    __shared__ float2 sW[(CH / 2) * CH];      // 32*64 float2 = 16 KB
    int tid = threadIdx.x;
    for (int i = tid; i < (CH / 2) * CH; i += 256) sW[i] = WpPK[i];
    __syncthreads();

    int lane = tid & 31;
    int wave = tid >> 5;
    int half = lane >> 4;                     // 0 for lanes 0-15, 1 for 16-31
    int l16  = lane & 15;
    int m0 = blockIdx.x * 32 + (wave >> 2) * 16;
    int n0 = (wave & 3) * 16;

    // A fragment row (clamped so EXEC stays all-1s; stores are guarded instead)
    int arow = m0 + l16;
    if (arow >= nrows) arow = nrows - 1;
    const float* aptr = agg + (size_t)arow * CH;

    float bias = bp[n0 + l16];                // N = l16 for every C/D register
    v8f c;
    #pragma unroll
    for (int v = 0; v < 8; ++v) c[v] = bias;

    #pragma unroll
    for (int k = 0; k < 16; ++k) {
        // ISA A-layout 16x4 f32: lanes 0-15 hold K={kb,kb+1}; lanes 16-31 K={kb+2,kb+3}
        int kb = 4 * k + 2 * half;
        float2 av = *(const float2*)(aptr + kb);
        v2f a; a.x = av.x; a.y = av.y;
        // B-layout mirrors A: lane holds column n0+l16, K pair {kb, kb+1}
        float2 bv = sW[(2 * k + half) * CH + n0 + l16];
        v2f b; b.x = bv.x; b.y = bv.y;
        c = __builtin_amdgcn_wmma_f32_16x16x4_f32(
                /*neg_a=*/false, a, /*neg_b=*/false, b,
                /*c_mod=*/(short)0, c, /*reuse_a=*/false, /*reuse_b=*/false);
    }

    #pragma unroll
    for (int v = 0; v < 8; ++v) {
        int row = m0 + v + 8 * half;          // D layout: VGPR v -> M = v + 8*half
        if (row < nrows) out[(size_t)row * CH + n0 + l16] = fmaxf(c[v], 0.0f);
    }
}

extern "C" void kernel_launch(void* const* d_in, const int* in_sizes, int n_in,
                              void* d_out, int out_size, void* d_ws, size_t ws_size,
                              hipStream_t stream) {
    const float* x       = (const float*)d_in[0];
    const int*   sources = (const int*)d_in[1];
    const int*   targets = (const int*)d_in[2];
    const float* gamma   = (const float*)d_in[3];
    const float* beta    = (const float*)d_in[4];
    const float* W       = (const float*)d_in[5];
    const float* b       = (const float*)d_in[6];
    float* out = (float*)d_out;

    int nnodes = in_sizes[0] / CH;
    int nedges = in_sizes[1];

    // workspace layout (floats)
    float*  sum       = (float*)d_ws;                       // nnodes*64 (becomes agg)
    float*  cnt       = sum + (size_t)nnodes * CH;          // nnodes
    float*  chanSum   = cnt + nnodes;                       // 64
    float*  chanSumSq = chanSum + CH;                       // 64
    float2* WpPK      = (float2*)(chanSumSq + CH);          // 32*64 float2 (8B aligned)
    float*  bp        = (float*)(WpPK + (CH / 2) * CH);     // 64

    long long nzero = (long long)nnodes * CH + nnodes + 2 * CH;
    zero_kernel<<<2048, 256, 0, stream>>>(sum, nzero);

    long long sthreads = (long long)nedges * CH;
    int sblocks = (int)((sthreads + 255) / 256);
    scatter_kernel<<<sblocks, 256, 0, stream>>>(x, sources, targets, sum, cnt, nedges);

    stats_kernel<<<512, 256, 0, stream>>>(sum, cnt, chanSum, chanSumSq, nnodes);

    prep_kernel<<<1, CH, 0, stream>>>(gamma, beta, W, b, chanSum, chanSumSq,
                                      WpPK, bp, 1.0f / (float)nnodes);

    int gblocks = (nnodes + 31) / 32;
    gemm_kernel<<<gblocks, 256, 0, stream>>>(sum, WpPK, bp, out, nnodes);
}